// LocalConvolutionMixMerge_70909910057703
// MI455X (gfx1250) — compile-verified
//
#include <hip/hip_runtime.h>

typedef float v2f __attribute__((ext_vector_type(2)));
typedef float v8f __attribute__((ext_vector_type(8)));

#define IMG_H 64
#define IMG_W 64
#define C_TOT 256
#define WC_N  32
#define G_N   8
#define T1_N  9    // 3x3
#define T2_N  25   // 5x5
#define NCH_W 1088 // 32*(9+25) weight planes
#define HW    (IMG_H * IMG_W)

// Select component r (0..7) of an 8-wide accumulator: 3-level cndmask tree.
__device__ __forceinline__ float pick8(v8f a, int r) {
    float s0 = (r & 1) ? a[1] : a[0];
    float s1 = (r & 1) ? a[3] : a[2];
    float s2 = (r & 1) ? a[5] : a[4];
    float s3 = (r & 1) ? a[7] : a[6];
    float t0 = (r & 2) ? s1 : s0;
    float t1 = (r & 2) ? s3 : s2;
    return (r & 4) ? t1 : t0;
}

__global__ __launch_bounds__(128)
void lconv_mixmerge_kernel(const float* __restrict__ x,
                           const float* __restrict__ wgt,
                           float* __restrict__ out) {
    const int tid  = threadIdx.x;
    const int wave = tid >> 5;
    const int lane = tid & 31;

    // block = (n, wc, h)
    const int bid = blockIdx.x;
    const int n   = bid / (WC_N * IMG_H);
    const int rem = bid - n * (WC_N * IMG_H);
    const int wc  = rem / IMG_H;
    const int h   = rem - wc * IMG_H;

    const int  w0   = wave * 16;       // wave owns 16 consecutive pixels
    const int  m    = lane & 15;       // M index (A) == N index (B) == pixel
    const bool hf   = (lane >> 4) != 0;// false: taps {4c,4c+1}, true: {4c+2,4c+3}
    const int  wpix = w0 + m;
    const int  wrow = h * IMG_W + wpix;

    // ---- per-tap patch offset + mask; tap ids are loop constants so all the
    //      /3 %3 /5 %5 math folds to immediates; `hf` select is one cndmask.
    auto prep_tap = [&](int t_a, int ksz, int pad, int T,
                        int& off, float& msk) {
        const int t_b  = t_a + 2;
        const int dy_a = t_a / ksz - pad, dx_a = t_a % ksz - pad;
        const int dy_b = t_b / ksz - pad, dx_b = t_b % ksz - pad;
        const int dy   = hf ? dy_b : dy_a;
        const int dx   = hf ? dx_b : dx_a;
        const bool tin = hf ? (t_b < T) : (t_a < T);
        const int hh = h + dy, ww = wpix + dx;
        const bool ok = tin && ((unsigned)hh < IMG_H) && ((unsigned)ww < IMG_W);
        off = ok ? (hh * IMG_W + ww) : 0;   // clamped: always a legal address
        msk = ok ? 1.0f : 0.0f;
    };

    // ---- B operands: per-pixel weights, loaded once, reused across g ------
    const float* wbase0 = wgt + (size_t)n * NCH_W * HW;

    auto load_w = [&](int t_a, int T, int planebase) -> float {
        const int t_b  = t_a + 2;
        const int ts_a = (t_a < T) ? t_a : (T - 1);   // clamp: legal address
        const int ts_b = (t_b < T) ? t_b : (T - 1);
        const int t    = hf ? ts_b : ts_a;
        const bool tin = hf ? (t_b < T) : (t_a < T);
        const float v  = wbase0[(planebase + t) * HW + wrow];
        return tin ? v : 0.0f;
    };

    v2f B1[3], B2[7];
#pragma unroll
    for (int c4 = 0; c4 < 3; ++c4) {
        B1[c4][0] = load_w(4 * c4 + 0, T1_N, wc * T1_N);
        B1[c4][1] = load_w(4 * c4 + 1, T1_N, wc * T1_N);
    }
#pragma unroll
    for (int c4 = 0; c4 < 7; ++c4) {
        B2[c4][0] = load_w(4 * c4 + 0, T2_N, WC_N * T1_N + wc * T2_N);
        B2[c4][1] = load_w(4 * c4 + 1, T2_N, WC_N * T1_N + wc * T2_N);
    }

    // ---- patch offsets / masks, invariant across g ------------------------
    int   off1[3][2], off2[7][2];
    float msk1[3][2], msk2[7][2];
#pragma unroll
    for (int c4 = 0; c4 < 3; ++c4) {
        prep_tap(4 * c4 + 0, 3, 1, T1_N, off1[c4][0], msk1[c4][0]);
        prep_tap(4 * c4 + 1, 3, 1, T1_N, off1[c4][1], msk1[c4][1]);
    }
#pragma unroll
    for (int c4 = 0; c4 < 7; ++c4) {
        prep_tap(4 * c4 + 0, 5, 2, T2_N, off2[c4][0], msk2[c4][0]);
        prep_tap(4 * c4 + 1, 5, 2, T2_N, off2[c4][1], msk2[c4][1]);
    }

    // diagonal holder lanes: 0..7 -> pixels 0..7, 24..31 -> pixels 8..15
    const int  r    = lane & 7;
    const bool have = (lane < 8) || (lane >= 24);
    const int  mp   = (lane < 8) ? lane : (lane - 16);

    // ---- per channel-group: fp32 WMMA over tap chunks ---------------------
    float d1v[G_N], d2v[G_N];
#pragma unroll
    for (int g = 0; g < G_N; ++g) {
        const int c = g * WC_N + wc;
        const float* xbase = x + (size_t)(n * C_TOT + c) * HW;

        v8f acc1 = {0.f, 0.f, 0.f, 0.f, 0.f, 0.f, 0.f, 0.f};
#pragma unroll
        for (int c4 = 0; c4 < 3; ++c4) {
            v2f A;
            A[0] = xbase[off1[c4][0]] * msk1[c4][0];   // unconditional load
            A[1] = xbase[off1[c4][1]] * msk1[c4][1];
            acc1 = __builtin_amdgcn_wmma_f32_16x16x4_f32(
                false, A, false, B1[c4], (short)0, acc1, false, false);
        }

        v8f acc2 = {0.f, 0.f, 0.f, 0.f, 0.f, 0.f, 0.f, 0.f};
#pragma unroll
        for (int c4 = 0; c4 < 7; ++c4) {
            v2f A;
            A[0] = xbase[off2[c4][0]] * msk2[c4][0];
            A[1] = xbase[off2[c4][1]] * msk2[c4][1];
            acc2 = __builtin_amdgcn_wmma_f32_16x16x4_f32(
                false, A, false, B2[c4], (short)0, acc2, false, false);
        }

        d1v[g] = pick8(acc1, r);
        d2v[g] = pick8(acc2, r);
    }

    // ---- single predicated epilogue: 16 contiguous-segment stores ---------
    if (have) {
        // out layout: [N, kernel=2, heads=1, C, H, W]
        const int ob = (n * 2 * C_TOT + wc) * HW + h * IMG_W + w0 + mp;
#pragma unroll
        for (int g = 0; g < G_N; ++g) {
            out[ob + g * WC_N * HW]               = d1v[g];
            out[ob + (C_TOT + g * WC_N) * HW]     = d2v[g];
        }
    }
}

extern "C" void kernel_launch(void* const* d_in, const int* in_sizes, int n_in,
                              void* d_out, int out_size, void* d_ws, size_t ws_size,
                              hipStream_t stream) {
    const float* x   = (const float*)d_in[0];   // [2, 256, 64, 64]
    const float* wgt = (const float*)d_in[1];   // [2, 1088, 64, 64]
    float* out = (float*)d_out;                 // [2, 2, 1, 256, 64, 64]

    const int blocks = 2 * WC_N * IMG_H;        // n * wc * h = 4096
    lconv_mixmerge_kernel<<<blocks, 128, 0, stream>>>(x, wgt, out);
}